// SparseGCNConv_81819126989160
// MI455X (gfx1250) — compile-verified
//
#include <hip/hip_runtime.h>

#define N_NODES   100000
#define N_EDGES   1600000
#define IN_FEATS  64
#define OUT_FEATS 64
#define EPW       256      // edges per wave in SpMM phase
#define LDS_STRIDE 68      // 64 + 4 pad dwords -> bank = (4n + k) % 64, conflict-free

typedef __attribute__((ext_vector_type(2))) float v2f;
typedef __attribute__((ext_vector_type(8))) float v8f;

// ---------------------------------------------------------------------------
// Phase 0: zero the aggregation buffer (d_out is poisoned with 0xAA)
// ---------------------------------------------------------------------------
__global__ void gcn_zero_kernel(float4* __restrict__ out, int n4) {
    int i = blockIdx.x * blockDim.x + threadIdx.x;
    if (i < n4) out[i] = make_float4(0.f, 0.f, 0.f, 0.f);
}

// ---------------------------------------------------------------------------
// Phase 1: SpMM  agg[r] += adj_vals[e] * features[col[e]]   (row sorted)
// One wave per 256 contiguous edges; 32 lanes each own a float2 feature slice
// (features 2*lane, 2*lane+1). Accumulate in registers while row id stays
// constant; flush with global atomic f32 adds only at segment boundaries.
// ---------------------------------------------------------------------------
__global__ void gcn_spmm_kernel(const float* __restrict__ features,
                                const float* __restrict__ vals,
                                const int*   __restrict__ row,
                                const int*   __restrict__ col,
                                float*       __restrict__ agg) {
    const int lane = threadIdx.x & 31;
    const int wave = blockIdx.x * (blockDim.x >> 5) + (threadIdx.x >> 5);
    int e0 = wave * EPW;
    if (e0 >= N_EDGES) return;
    int e1 = e0 + EPW;
    if (e1 > N_EDGES) e1 = N_EDGES;

    const float2* f2 = (const float2*)features;   // [N_NODES][32] float2
    float accx = 0.f, accy = 0.f;
    int cur = row[e0];                            // uniform across wave

    for (int e = e0; e < e1; e += 32) {
        const int nv = min(32, e1 - e);
        int   r = (lane < nv) ? row[e + lane]  : 0;
        int   c = (lane < nv) ? col[e + lane]  : 0;
        float v = (lane < nv) ? vals[e + lane] : 0.f;
        for (int j = 0; j < nv; ++j) {
            int   rj = __shfl(r, j, 32);          // uniform broadcast
            int   cj = __shfl(c, j, 32);
            float vj = __shfl(v, j, 32);
            if (rj != cur) {                      // segment boundary: flush
                atomicAdd(&agg[cur * 64 + 2 * lane],     accx);
                atomicAdd(&agg[cur * 64 + 2 * lane + 1], accy);
                accx = 0.f; accy = 0.f;
                cur = rj;
            }
            float2 f = f2[(size_t)cj * 32 + lane]; // coalesced b64 gather (L2 hit)
            accx += vj * f.x;
            accy += vj * f.y;
        }
    }
    atomicAdd(&agg[cur * 64 + 2 * lane],     accx);
    atomicAdd(&agg[cur * 64 + 2 * lane + 1], accy);
}

// ---------------------------------------------------------------------------
// Phase 2: in-place GEMM  out = agg @ W + bias  via V_WMMA_F32_16X16X4_F32.
// Each wave: one 16-row tile x 64 cols = 4 (16x16) accumulators, K=64 in 16
// steps of 4. W transposed into padded LDS so B fragments are bank-conflict-
// free ds_load_b64. C initialized from bias (free bias add).
// In-place is safe: a wave reads all of rows [m0,m0+16) before storing them,
// and row tiles are disjoint across waves.
// ---------------------------------------------------------------------------
__global__ void gcn_gemm_wmma_kernel(const float* __restrict__ weight,  // [64][64] W[k][n]
                                     const float* __restrict__ bias,    // [64]
                                     float*       __restrict__ inout) { // [N][64]
    __shared__ float Wt[64 * LDS_STRIDE];         // Wt[n][k]
    const int tid = threadIdx.x;

    #pragma unroll
    for (int i = 0; i < 16; ++i) {                // 256 thr x 16 = 4096 elems
        int idx = tid + i * 256;
        int k = idx >> 6, n = idx & 63;
        Wt[n * LDS_STRIDE + k] = weight[idx];
    }
    __syncthreads();

    const int wave = (blockIdx.x << 3) + (tid >> 5);
    if (wave >= N_NODES / 16) return;             // 100000 = 6250 * 16 exactly

    const int lane = tid & 31;
    const int m0   = wave * 16;
    const int n    = lane & 15;                   // B/C column, A row (mod 16)
    const int kb   = (lane >> 4) << 1;            // K sub-offset: 0 or 2
    const int half = lane >> 4;                   // C row offset: 0 or 8

    v8f c[4];
    #pragma unroll
    for (int nt = 0; nt < 4; ++nt) {
        float b = bias[nt * 16 + n];
        #pragma unroll
        for (int j = 0; j < 8; ++j) c[nt][j] = b;
    }

    const float* arow = inout + (size_t)(m0 + n) * 64 + kb;  // A: row m=lane&15
    #pragma unroll
    for (int k0 = 0; k0 < 64; k0 += 4) {
        v2f a = *(const v2f*)(arow + k0);                    // A[m][k0+kb .. +1]
        #pragma unroll
        for (int nt = 0; nt < 4; ++nt) {
            // B[k0+kb][nt*16+n], B[k0+kb+1][...] contiguous in transposed LDS
            v2f b = *(const v2f*)&Wt[(nt * 16 + n) * LDS_STRIDE + k0 + kb];
            c[nt] = __builtin_amdgcn_wmma_f32_16x16x4_f32(
                        false, a, false, b, (short)0, c[nt], false, false);
        }
    }

    #pragma unroll
    for (int nt = 0; nt < 4; ++nt) {
        #pragma unroll
        for (int j = 0; j < 8; ++j) {
            inout[(size_t)(m0 + j + 8 * half) * 64 + nt * 16 + n] = c[nt][j];
        }
    }
}

// ---------------------------------------------------------------------------
extern "C" void kernel_launch(void* const* d_in, const int* in_sizes, int n_in,
                              void* d_out, int out_size, void* d_ws, size_t ws_size,
                              hipStream_t stream) {
    const float* features = (const float*)d_in[0];
    const float* adj_vals = (const float*)d_in[1];
    const float* weight   = (const float*)d_in[2];
    const float* bias     = (const float*)d_in[3];
    const int*   row      = (const int*)d_in[4];
    const int*   col      = (const int*)d_in[5];
    float*       out      = (float*)d_out;

    // Phase 0: zero agg (== d_out)
    const int n4 = (N_NODES * OUT_FEATS) / 4;
    gcn_zero_kernel<<<(n4 + 255) / 256, 256, 0, stream>>>((float4*)out, n4);

    // Phase 1: SpMM into d_out
    const int spmm_waves  = (N_EDGES + EPW - 1) / EPW;      // 6250
    const int spmm_blocks = (spmm_waves + 7) / 8;           // 8 waves / block
    gcn_spmm_kernel<<<spmm_blocks, 256, 0, stream>>>(features, adj_vals, row, col, out);

    // Phase 2: in-place WMMA GEMM + bias
    const int gemm_waves  = N_NODES / 16;                   // 6250 row tiles
    const int gemm_blocks = (gemm_waves + 7) / 8;
    gcn_gemm_wmma_kernel<<<gemm_blocks, 256, 0, stream>>>(weight, bias, out);
}